// LowRankMaskedSynapse_78426102825457
// MI455X (gfx1250) — compile-verified
//
#include <hip/hip_runtime.h>

typedef __attribute__((ext_vector_type(2))) float v2f;
typedef __attribute__((ext_vector_type(8))) float v8f;

#define N_PRE  16384
#define N_POST 16384
#define KIDX   64
#define RANK   32
#define BATCH  32

// ---------------------------------------------------------------------------
// Pass 1: transpose x[BATCH][N_PRE] -> xT[N_PRE][BATCH] so the per-n gather of
// x "columns" becomes 128-byte contiguous row reads (L2-friendly).
// ---------------------------------------------------------------------------
__global__ void __launch_bounds__(1024)
transpose_x_kernel(const float* __restrict__ x, float* __restrict__ xT) {
    __shared__ float tile[32][33];
    const unsigned p0 = blockIdx.x * 32u;
    const unsigned tx = threadIdx.x;   // 0..31
    const unsigned ty = threadIdx.y;   // 0..31
    // coalesced read along N_PRE
    tile[ty][tx] = x[ty * (unsigned)N_PRE + p0 + tx];
    __syncthreads();
    // coalesced write of 128B rows of xT
    xT[(p0 + ty) * (unsigned)BATCH + tx] = tile[tx][ty];
}

// ---------------------------------------------------------------------------
// Pass 2: one wave32 per output column n.
//   G [64x32]  : G[k,b]  = xT[idx[n,k]][b]   (gathered x columns)
//   Vg[64x32]  : Vg[k,r] = V[idx[n,k]][r]    (gathered V rows)
//   T = G^T * Vg  (32x32x64 f32 GEMM via V_WMMA_F32_16X16X4_F32, 2x2 tiles)
//   y[b,n] = sum_r T[b,r] * U[n,r]           (shuffle reduction over 16 lanes)
//
// All gathers use a single 32-bit element offset per gathered row
// (off = p*32 + m), shared by the xT and V loads (both have 32-float rows),
// so the compiler can emit SADDR-form global_load (SGPR base + VGPR off32)
// with the +16 lane-group folded into the instruction IOFFSET.
// ---------------------------------------------------------------------------
__global__ void __launch_bounds__(256)
LowRankMaskedSynapse_kernel(const float* __restrict__ xT,
                            const float* __restrict__ U,
                            const float* __restrict__ V,
                            const int*   __restrict__ idx,
                            float*       __restrict__ y) {
    const unsigned lane = threadIdx.x & 31u;
    const unsigned wave = threadIdx.x >> 5;
    const unsigned n    = blockIdx.x * 8u + wave;   // grid sized so n < N_POST
    const unsigned h    = lane >> 4;                // half-wave
    const unsigned m    = lane & 15u;               // row/col within 16-tile

    const int* __restrict__ idx_n = idx + n * (unsigned)KIDX;

    v8f acc00 = {}, acc01 = {}, acc10 = {}, acc11 = {};

    // 16 K-steps of 4; per step this wave consumes rows 4kk..4kk+3 of G/Vg.
    // ISA operand layout (f32 16x4 A and 4x16 B): lane (h,m), VGPR v holds
    // K = kb + v (kb = 4kk + 2h), M/N = m  -> both operands index identically.
    #pragma unroll
    for (int kk = 0; kk < 16; ++kk) {
        const unsigned kb = (unsigned)kk * 4u + 2u * h;
        const int2 p = *(const int2*)(idx_n + kb);   // global_load_b64

        // 32-bit element offsets (max 16384*32 < 2^21, sign bit clear)
        const unsigned off0 = ((unsigned)p.x << 5) + m;
        const unsigned off1 = ((unsigned)p.y << 5) + m;

        v2f a0, a1, b0, b1;
        a0.x = xT[off0];        a0.y = xT[off1];        // batch rows 0..15
        a1.x = xT[off0 + 16u];  a1.y = xT[off1 + 16u];  // batch rows 16..31
        b0.x = V[off0];         b0.y = V[off1];         // r = 0..15
        b1.x = V[off0 + 16u];   b1.y = V[off1 + 16u];   // r = 16..31

        acc00 = __builtin_amdgcn_wmma_f32_16x16x4_f32(false, a0, false, b0,
                                                      (short)0, acc00, false, false);
        acc01 = __builtin_amdgcn_wmma_f32_16x16x4_f32(false, a0, false, b1,
                                                      (short)0, acc01, false, false);
        acc10 = __builtin_amdgcn_wmma_f32_16x16x4_f32(false, a1, false, b0,
                                                      (short)0, acc10, false, false);
        acc11 = __builtin_amdgcn_wmma_f32_16x16x4_f32(false, a1, false, b1,
                                                      (short)0, acc11, false, false);
    }

    // Final contraction with U[n,:].  C/D layout: lane (h,m), VGPR v holds
    // T[b = mtile*16 + 8h + v, r = ntile*16 + m].
    const unsigned un = n * (unsigned)RANK + m;
    const float u0 = U[un];
    const float u1 = U[un + 16u];

    float out0[8], out1[8];
    #pragma unroll
    for (int v = 0; v < 8; ++v) {
        float s0 = acc00[v] * u0 + acc01[v] * u1;   // b = 8h + v
        float s1 = acc10[v] * u0 + acc11[v] * u1;   // b = 16 + 8h + v
        #pragma unroll
        for (int mask = 8; mask >= 1; mask >>= 1) { // reduce over the 16 lanes of this half
            s0 += __shfl_xor(s0, mask, 32);
            s1 += __shfl_xor(s1, mask, 32);
        }
        out0[v] = s0;
        out1[v] = s1;
    }

    if (m == 0u) {  // lane 0 and lane 16 each hold 16 of the 32 batch results
        #pragma unroll
        for (int v = 0; v < 8; ++v) {
            y[(8u * h + (unsigned)v)       * (unsigned)N_POST + n] = out0[v];
            y[(16u + 8u * h + (unsigned)v) * (unsigned)N_POST + n] = out1[v];
        }
    }
}

// ---------------------------------------------------------------------------
extern "C" void kernel_launch(void* const* d_in, const int* in_sizes, int n_in,
                              void* d_out, int out_size, void* d_ws, size_t ws_size,
                              hipStream_t stream) {
    const float* x   = (const float*)d_in[0];   // [BATCH, N_PRE]
    const float* U   = (const float*)d_in[1];   // [N_POST, RANK]
    const float* V   = (const float*)d_in[2];   // [N_PRE, RANK]
    const int*   idx = (const int*)  d_in[3];   // [N_POST, KIDX]
    float*       y   = (float*)d_out;           // [BATCH, N_POST]
    float*       xT  = (float*)d_ws;            // [N_PRE, BATCH] scratch (2 MB)

    dim3 tb(32, 32);
    transpose_x_kernel<<<N_PRE / 32, tb, 0, stream>>>(x, xT);

    LowRankMaskedSynapse_kernel<<<N_POST / 8, 256, 0, stream>>>(xT, U, V, idx, y);
}